// myLayerRotation_8727373546134
// MI455X (gfx1250) — compile-verified
//
#include <hip/hip_runtime.h>

// Shapes from the reference: x = (256, 3, 65536) f32, weight = (3,3) f32,
// out = (256, 3, 65536) f32.
#define BATCH     256
#define NCOL      65536
#define ROWSTRIDE NCOL            // stride between j-rows within a batch
#define BSTRIDE   (3 * NCOL)      // stride between batches
#define PAIRS_PER_WAVE 4          // each pair = 2 x 16-column WMMA tiles = 32 cols
// pairs per batch = NCOL/32 = 2048; total pairs = 256*2048 = 524288
// waves = 524288/4 = 131072 -> 16384 blocks of 256 threads (8 waves)

typedef __attribute__((ext_vector_type(2))) float v2f;
typedef __attribute__((ext_vector_type(8))) float v8f;

__global__ __launch_bounds__(256) void w3x3_stream_wmma_kernel(
    const float* __restrict__ x, const float* __restrict__ w,
    float* __restrict__ out)
{
    const int lane  = threadIdx.x & 31;
    const int col16 = lane & 15;          // column within a 16-wide tile
    const bool hi   = lane >= 16;         // lane-half selects K range

    // ---- Load the 3x3 weight (uniform) and build the A fragment ----------
    // A is the 16x4 zero-padded W. ISA layout (32-bit A 16x4):
    //   VGPR0: lanes 0-15 -> (M=lane, K=0), lanes 16-31 -> (M=lane-16, K=2)
    //   VGPR1: lanes 0-15 -> (M=lane, K=1), lanes 16-31 -> (M=lane-16, K=3)
    const float w00 = w[0], w01 = w[1], w02 = w[2];
    const float w10 = w[3], w11 = w[4], w12 = w[5];
    const float w20 = w[6], w21 = w[7], w22 = w[8];

    float a0 = (col16 == 0) ? (hi ? w02 : w00)
             : (col16 == 1) ? (hi ? w12 : w10)
             : (col16 == 2) ? (hi ? w22 : w20) : 0.0f;      // K=0 or K=2
    float a1 = hi ? 0.0f                                     // K=3 column is pad
             : ((col16 == 0) ? w01
              : (col16 == 1) ? w11
              : (col16 == 2) ? w21 : 0.0f);                  // K=1
    v2f A;  A.x = a0;  A.y = a1;

    const long long gwave =
        (((long long)blockIdx.x * blockDim.x) + threadIdx.x) >> 5;
    const long long q0 = gwave * PAIRS_PER_WAVE;  // first tile-pair index

    const int bidx = col16 << 2;  // byte index for ds_bpermute (lane%16)*4

#pragma unroll
    for (int p = 0; p < PAIRS_PER_WAVE; ++p) {
        const long long q    = q0 + p;
        const int b          = (int)(q >> 11);      // q / 2048
        const int pin        = (int)(q & 2047);     // pair within batch
        const int cbase      = pin << 5;            // 32 columns per pair
        const float* __restrict__ xb = x   + (long long)b * BSTRIDE;
        float*       __restrict__ ob = out + (long long)b * BSTRIDE;

        // ---- Load B fragments for two adjacent 16-col tiles --------------
        // B is 4x16 (KxN): lane%16 = N, lane-half selects K in {0,2}/{1,3}.
        // VGPR0: lanes0-15 read row0, lanes16-31 read row2 (2x64B useful).
        // VGPR1: row1 (upper half is a redundant read of the same 64B,
        //         then zeroed for the K=3 pad row).
        const int rsel = hi ? (2 * ROWSTRIDE) : 0;
        const int c0 = cbase + col16;
        const int c1 = cbase + 16 + col16;

        v2f B0, B1;
        B0.x = xb[rsel + c0];
        B1.x = xb[rsel + c1];
        const float r1a = xb[ROWSTRIDE + c0];
        const float r1b = xb[ROWSTRIDE + c1];
        B0.y = hi ? 0.0f : r1a;
        B1.y = hi ? 0.0f : r1b;

        // ---- D = A x B (+0) : v_wmma_f32_16x16x4_f32 ---------------------
        v8f C0 = {}; v8f C1 = {};
        v8f D0 = __builtin_amdgcn_wmma_f32_16x16x4_f32(
            false, A, false, B0, (short)0, C0, false, false);
        v8f D1 = __builtin_amdgcn_wmma_f32_16x16x4_f32(
            false, A, false, B1, (short)0, C1, false, false);

        // ---- Merge the two half-wave results -> full 128B stores ---------
        // D layout: VGPR m, lanes 0-15 hold (M=m, N=lane). We need M=0..2.
        // Lanes 16-31 take tile1's column (lane-16) via ds_bpermute.
#pragma unroll
        for (int m = 0; m < 3; ++m) {
            const float lo = D0[m];
            const int permed = __builtin_amdgcn_ds_bpermute(
                bidx, __builtin_bit_cast(int, D1[m]));
            const float hiv = __builtin_bit_cast(float, permed);
            const float v = hi ? hiv : lo;
            ob[(long long)m * ROWSTRIDE + cbase + lane] = v;
        }
    }
}

extern "C" void kernel_launch(void* const* d_in, const int* in_sizes, int n_in,
                              void* d_out, int out_size, void* d_ws, size_t ws_size,
                              hipStream_t stream) {
    const float* x = (const float*)d_in[0];   // (256, 3, 65536) f32
    const float* w = (const float*)d_in[1];   // (3, 3) f32
    float* out = (float*)d_out;               // (256, 3, 65536) f32

    // 256 batches * 2048 pairs / 4 pairs-per-wave = 131072 waves
    // = 4,194,304 threads = 16384 blocks of 256.
    dim3 grid(16384), block(256);
    hipLaunchKernelGGL(w3x3_stream_wmma_kernel, grid, block, 0, stream,
                       x, w, out);
}